// HyperGraphNet_44109314130696
// MI455X (gfx1250) — compile-verified
//
#include <hip/hip_runtime.h>

typedef __attribute__((ext_vector_type(2))) float v2f;
typedef __attribute__((ext_vector_type(8))) float v8f;

#define D 128
#define N_ATOMS 100000
#define N_BONDS 200000
#define N_ANGLES 300000
#define N_TORSIONS 400000
#define N_MOLS 10000
#define REPEAT 3

#define ROWS 64          // rows (M) per block
#define LDSP 68          // padded LDS row stride in dwords: 16B-aligned, bank-spread

// ---------------------------------------------------------------------------
// Core: one block (256 thr = 8 wave32) computes a 64-row x 128-col tile of
// acc = A(64xK) * W(KxD).  A rows are gathered via per-segment base pointer
// (kernel arg, stays addrspace-global) + row index from LDS.  A and W are
// staged through LDS in 64-wide K chunks; W chunk is stored transposed so
// both fragments are single ds_load_b64 matching the 16x4 / 4x16 WMMA lane
// layout (lanes 0-15: K=k,k+1; lanes 16-31: K=k+2,k+3).
// Each wave owns a 2x2 grid of 16x16 V_WMMA_F32_16X16X4_F32 accumulators:
//   wave w -> M-tiles {2*(w>>2), 2*(w>>2)+1}, N-tiles {2*(w&3), 2*(w&3)+1}
// => 4 ds_load_b64 feed 4 WMMAs (1.0 ds/WMMA), 4 independent acc chains.
// ---------------------------------------------------------------------------
template<int K>
__device__ __forceinline__ void gemm_tile(const float* __restrict__ W,
                                          const float* const* segBase, // [K/128]
                                          const int* sIdx,             // shared [(K/128)*ROWS]
                                          float* As,                   // ROWS*LDSP
                                          float* BsT,                  // 128*LDSP
                                          v8f acc[2][2])
{
    const int t    = threadIdx.x;
    const int lane = t & 31;
    const int wave = t >> 5;
    const int m    = lane & 15;
    const int koff = (lane >> 4) * 2;      // hi half-wave holds K+2,K+3
    const int mt0  = (wave >> 2) * 2;      // first M-tile of this wave
    const int ct0  = (wave & 3) * 2;       // first N-tile of this wave

#pragma unroll
    for (int k0 = 0; k0 < K; k0 += 64) {
        const int seg = k0 >> 7;           // constant after unroll
        const int off = k0 & 127;
        __syncthreads();                   // protect previous chunk reads (+ sIdx init)

        // ---- stage A chunk: ROWS x 64, float4 per thread x4 ---------------
#pragma unroll
        for (int i = 0; i < (ROWS * 16) / 256; ++i) {
            const int q   = t + i * 256;           // quad id
            const int row = q >> 4;                // 0..ROWS-1
            const int col = (q & 15) * 4;          // 0..60
            const int eoff = sIdx[seg * ROWS + row] * D + off + col;  // < 2^27
            *(float4*)&As[row * LDSP + col] = *(const float4*)(segBase[seg] + eoff);
        }
        // ---- stage B chunk transposed: W[k0..k0+63][:] -> BsT[n][k] -------
        {
            const float* wsrc = W + (size_t)k0 * D;   // contiguous 64*128 floats
#pragma unroll
            for (int i = 0; i < 8; ++i) {
                const int e = (t + i * 256) * 4;
                const float4 w4 = *(const float4*)(wsrc + e);
                const int k = e >> 7;
                const int n = e & 127;
                BsT[(n + 0) * LDSP + k] = w4.x;
                BsT[(n + 1) * LDSP + k] = w4.y;
                BsT[(n + 2) * LDSP + k] = w4.z;
                BsT[(n + 3) * LDSP + k] = w4.w;
            }
            if (k0 + 64 < K) __builtin_prefetch(wsrc + 64 * D, 0, 0); // global_prefetch_b8
        }
        __syncthreads();

        // ---- 16 k-steps x 4 WMMAs over this chunk -------------------------
#pragma unroll
        for (int kk = 0; kk < 64; kk += 4) {
            const v2f a0 = *(const v2f*)&As[(mt0 * 16 + m) * LDSP + kk + koff];
            const v2f a1 = *(const v2f*)&As[((mt0 + 1) * 16 + m) * LDSP + kk + koff];
            const v2f b0 = *(const v2f*)&BsT[(ct0 * 16 + m) * LDSP + kk + koff];
            const v2f b1 = *(const v2f*)&BsT[((ct0 + 1) * 16 + m) * LDSP + kk + koff];
            acc[0][0] = __builtin_amdgcn_wmma_f32_16x16x4_f32(false, a0, false, b0, (short)0, acc[0][0], false, false);
            acc[0][1] = __builtin_amdgcn_wmma_f32_16x16x4_f32(false, a0, false, b1, (short)0, acc[0][1], false, false);
            acc[1][0] = __builtin_amdgcn_wmma_f32_16x16x4_f32(false, a1, false, b0, (short)0, acc[1][0], false, false);
            acc[1][1] = __builtin_amdgcn_wmma_f32_16x16x4_f32(false, a1, false, b1, (short)0, acc[1][1], false, false);
        }
    }
}

// Epilogue indexing: C/D layout -> lane l, vgpr j of tile (mt,ct) holds
// local row = mt*16 + j + 8*(l>=16), col = ct*16 + (l&15).
#define EPILOGUE_VARS()                                   \
    const int lane_ = threadIdx.x & 31;                   \
    const int wave_ = threadIdx.x >> 5;                   \
    const int mt0_  = (wave_ >> 2) * 2;                   \
    const int ct0_  = (wave_ & 3) * 2;                    \
    const int nlo_  = lane_ & 15;                         \
    const int rhalf_ = (lane_ >> 4) << 3;

#define ZACC(acc) { v8f z = {0.f,0.f,0.f,0.f,0.f,0.f,0.f,0.f}; \
    acc[0][0]=z; acc[0][1]=z; acc[1][0]=z; acc[1][1]=z; }

// ---------------------------------------------------------------------------
// Edge MLP: h_e = relu([h_v[b0] | h_v[b1] | u[mol]] @ W_e + b_e)
// fused scatter: e_bar[b0]+=h_e, e_bar[b1]+=h_e, e_u[mol]+=h_e
// ---------------------------------------------------------------------------
__global__ __launch_bounds__(256)
void edge_kernel(const float* __restrict__ hv, const float* __restrict__ u,
                 const int* __restrict__ bond_idxs, const int* __restrict__ atom_mol,
                 const float* __restrict__ We, const float* __restrict__ be,
                 float* __restrict__ e_bar, float* __restrict__ e_u)
{
    __shared__ float As[ROWS * LDSP];
    __shared__ float BsT[128 * LDSP];
    __shared__ int sIdx[3 * ROWS];

    const int t = threadIdx.x;
    const int r0 = blockIdx.x * ROWS;
    if (t < ROWS) {
        const int r = min(r0 + t, N_BONDS - 1);
        const int b0 = bond_idxs[2 * r];
        const int b1 = bond_idxs[2 * r + 1];
        sIdx[t]            = b0;
        sIdx[ROWS + t]     = b1;
        sIdx[2 * ROWS + t] = atom_mol[b0];
    }
    const float* segBase[3] = { hv, hv, u };
    v8f acc[2][2]; ZACC(acc);
    gemm_tile<384>(We, segBase, sIdx, As, BsT, acc);

    EPILOGUE_VARS();
#pragma unroll
    for (int ci = 0; ci < 2; ++ci) {
        const int col = (ct0_ + ci) * 16 + nlo_;
        const float bb = be[col];
#pragma unroll
        for (int mi = 0; mi < 2; ++mi)
#pragma unroll
            for (int j = 0; j < 8; ++j) {
                const int lrow = (mt0_ + mi) * 16 + j + rhalf_;
                if (r0 + lrow < N_BONDS) {
                    const float v = fmaxf(acc[mi][ci][j] + bb, 0.f);
                    atomicAdd(&e_bar[(size_t)sIdx[lrow] * D + col], v);
                    atomicAdd(&e_bar[(size_t)sIdx[ROWS + lrow] * D + col], v);
                    atomicAdd(&e_u[(size_t)sIdx[2 * ROWS + lrow] * D + col], v);
                }
            }
    }
}

// ---------------------------------------------------------------------------
// Node MLP: h_v = relu([h_v | e_bar | u[mol]] @ W_v + b_v); v_u[mol]+=h_v
// In-place h_v safe: each block reads only its own rows before writing them.
// ---------------------------------------------------------------------------
__global__ __launch_bounds__(256)
void node_kernel(float* __restrict__ hv, const float* __restrict__ e_bar,
                 const float* __restrict__ u, const int* __restrict__ atom_mol,
                 const float* __restrict__ Wv, const float* __restrict__ bvec,
                 float* __restrict__ v_u)
{
    __shared__ float As[ROWS * LDSP];
    __shared__ float BsT[128 * LDSP];
    __shared__ int sIdx[3 * ROWS];

    const int t = threadIdx.x;
    const int r0 = blockIdx.x * ROWS;
    if (t < ROWS) {
        const int r = min(r0 + t, N_ATOMS - 1);
        sIdx[t]            = r;
        sIdx[ROWS + t]     = r;
        sIdx[2 * ROWS + t] = atom_mol[r];
    }
    const float* segBase[3] = { hv, e_bar, u };
    v8f acc[2][2]; ZACC(acc);
    gemm_tile<384>(Wv, segBase, sIdx, As, BsT, acc);

    EPILOGUE_VARS();
#pragma unroll
    for (int ci = 0; ci < 2; ++ci) {
        const int col = (ct0_ + ci) * 16 + nlo_;
        const float bb = bvec[col];
#pragma unroll
        for (int mi = 0; mi < 2; ++mi)
#pragma unroll
            for (int j = 0; j < 8; ++j) {
                const int lrow = (mt0_ + mi) * 16 + j + rhalf_;
                if (r0 + lrow < N_ATOMS) {
                    const float v = fmaxf(acc[mi][ci][j] + bb, 0.f);
                    hv[(size_t)(r0 + lrow) * D + col] = v;
                    atomicAdd(&v_u[(size_t)sIdx[2 * ROWS + lrow] * D + col], v);
                }
            }
    }
}

// ---------------------------------------------------------------------------
// Angle MLP: a_u[mol] += relu([hv[a0]|hv[a1]|hv[a2]|u[mol]] @ W_a + b_a)
// ---------------------------------------------------------------------------
__global__ __launch_bounds__(256)
void angle_kernel(const float* __restrict__ hv, const float* __restrict__ u,
                  const int* __restrict__ angle_idxs, const int* __restrict__ atom_mol,
                  const float* __restrict__ Wa, const float* __restrict__ ba,
                  float* __restrict__ a_u)
{
    __shared__ float As[ROWS * LDSP];
    __shared__ float BsT[128 * LDSP];
    __shared__ int sIdx[4 * ROWS];

    const int t = threadIdx.x;
    const int r0 = blockIdx.x * ROWS;
    if (t < ROWS) {
        const int r = min(r0 + t, N_ANGLES - 1);
        const int a0 = angle_idxs[3 * r];
        const int a1 = angle_idxs[3 * r + 1];
        const int a2 = angle_idxs[3 * r + 2];
        sIdx[t]            = a0;
        sIdx[ROWS + t]     = a1;
        sIdx[2 * ROWS + t] = a2;
        sIdx[3 * ROWS + t] = atom_mol[a1];
    }
    const float* segBase[4] = { hv, hv, hv, u };
    v8f acc[2][2]; ZACC(acc);
    gemm_tile<512>(Wa, segBase, sIdx, As, BsT, acc);

    EPILOGUE_VARS();
#pragma unroll
    for (int ci = 0; ci < 2; ++ci) {
        const int col = (ct0_ + ci) * 16 + nlo_;
        const float bb = ba[col];
#pragma unroll
        for (int mi = 0; mi < 2; ++mi)
#pragma unroll
            for (int j = 0; j < 8; ++j) {
                const int lrow = (mt0_ + mi) * 16 + j + rhalf_;
                if (r0 + lrow < N_ANGLES) {
                    const float v = fmaxf(acc[mi][ci][j] + bb, 0.f);
                    atomicAdd(&a_u[(size_t)sIdx[3 * ROWS + lrow] * D + col], v);
                }
            }
    }
}

// ---------------------------------------------------------------------------
// Torsion MLP: t_u[mol] += relu([hv[t0..t3]|u[mol]] @ W_t + b_t)
// ---------------------------------------------------------------------------
__global__ __launch_bounds__(256)
void torsion_kernel(const float* __restrict__ hv, const float* __restrict__ u,
                    const int* __restrict__ torsion_idxs, const int* __restrict__ atom_mol,
                    const float* __restrict__ Wt, const float* __restrict__ bt,
                    float* __restrict__ t_u)
{
    __shared__ float As[ROWS * LDSP];
    __shared__ float BsT[128 * LDSP];
    __shared__ int sIdx[5 * ROWS];

    const int t = threadIdx.x;
    const int r0 = blockIdx.x * ROWS;
    if (t < ROWS) {
        const int r = min(r0 + t, N_TORSIONS - 1);
        const int i0 = torsion_idxs[4 * r];
        const int i1 = torsion_idxs[4 * r + 1];
        const int i2 = torsion_idxs[4 * r + 2];
        const int i3 = torsion_idxs[4 * r + 3];
        sIdx[t]            = i0;
        sIdx[ROWS + t]     = i1;
        sIdx[2 * ROWS + t] = i2;
        sIdx[3 * ROWS + t] = i3;
        sIdx[4 * ROWS + t] = atom_mol[i1];
    }
    const float* segBase[5] = { hv, hv, hv, hv, u };
    v8f acc[2][2]; ZACC(acc);
    gemm_tile<640>(Wt, segBase, sIdx, As, BsT, acc);

    EPILOGUE_VARS();
#pragma unroll
    for (int ci = 0; ci < 2; ++ci) {
        const int col = (ct0_ + ci) * 16 + nlo_;
        const float bb = bt[col];
#pragma unroll
        for (int mi = 0; mi < 2; ++mi)
#pragma unroll
            for (int j = 0; j < 8; ++j) {
                const int lrow = (mt0_ + mi) * 16 + j + rhalf_;
                if (r0 + lrow < N_TORSIONS) {
                    const float v = fmaxf(acc[mi][ci][j] + bb, 0.f);
                    atomicAdd(&t_u[(size_t)sIdx[4 * ROWS + lrow] * D + col], v);
                }
            }
    }
}

// ---------------------------------------------------------------------------
// Global-state MLP: u = relu([u | e_u | v_u | a_u | t_u] @ W_u + b_u), in place.
// ---------------------------------------------------------------------------
__global__ __launch_bounds__(256)
void u_kernel(float* __restrict__ u, const float* __restrict__ e_u,
              const float* __restrict__ v_u, const float* __restrict__ a_u,
              const float* __restrict__ t_u,
              const float* __restrict__ Wu, const float* __restrict__ bu)
{
    __shared__ float As[ROWS * LDSP];
    __shared__ float BsT[128 * LDSP];
    __shared__ int sIdx[5 * ROWS];

    const int t = threadIdx.x;
    const int r0 = blockIdx.x * ROWS;
    if (t < ROWS) {
        const int g = min(r0 + t, N_MOLS - 1);
        sIdx[t]            = g;
        sIdx[ROWS + t]     = g;
        sIdx[2 * ROWS + t] = g;
        sIdx[3 * ROWS + t] = g;
        sIdx[4 * ROWS + t] = g;
    }
    const float* segBase[5] = { u, e_u, v_u, a_u, t_u };
    v8f acc[2][2]; ZACC(acc);
    gemm_tile<640>(Wu, segBase, sIdx, As, BsT, acc);

    EPILOGUE_VARS();
#pragma unroll
    for (int ci = 0; ci < 2; ++ci) {
        const int col = (ct0_ + ci) * 16 + nlo_;
        const float bb = bu[col];
#pragma unroll
        for (int mi = 0; mi < 2; ++mi)
#pragma unroll
            for (int j = 0; j < 8; ++j) {
                const int lrow = (mt0_ + mi) * 16 + j + rhalf_;
                if (r0 + lrow < N_MOLS)
                    u[(size_t)(r0 + lrow) * D + col] = fmaxf(acc[mi][ci][j] + bb, 0.f);
            }
    }
}

// ---------------------------------------------------------------------------
// Utility kernels
// ---------------------------------------------------------------------------
__global__ void zero_kernel(float* __restrict__ p, long n4)
{
    const long i = (long)blockIdx.x * blockDim.x + threadIdx.x;
    if (i < n4) ((float4*)p)[i] = make_float4(0.f, 0.f, 0.f, 0.f);
}

__global__ void copy_kernel(float* __restrict__ dst, const float* __restrict__ src, long n4)
{
    const long i = (long)blockIdx.x * blockDim.x + threadIdx.x;
    if (i < n4) ((float4*)dst)[i] = ((const float4*)src)[i];
}

__global__ void init_hv_kernel(const int* __restrict__ atoms,
                               const float* __restrict__ embed,
                               float* __restrict__ hv)
{
    const long q = (long)blockIdx.x * blockDim.x + threadIdx.x;  // quad id
    const long a = q >> 5;
    const int  c = (int)(q & 31) << 2;
    if (a < N_ATOMS) {
        const int ty = atoms[a];
        *(float4*)&hv[a * D + c] = *(const float4*)&embed[(size_t)ty * D + c];
    }
}

static inline int div_up_l(long a, long b) { return (int)((a + b - 1) / b); }

// ---------------------------------------------------------------------------
extern "C" void kernel_launch(void* const* d_in, const int* in_sizes, int n_in,
                              void* d_out, int out_size, void* d_ws, size_t ws_size,
                              hipStream_t stream)
{
    (void)in_sizes; (void)n_in; (void)out_size; (void)ws_size;

    const int*   atoms        = (const int*)d_in[0];
    const int*   bond_idxs    = (const int*)d_in[1];
    const int*   angle_idxs   = (const int*)d_in[2];
    const int*   torsion_idxs = (const int*)d_in[3];
    const int*   atom_mol     = (const int*)d_in[4];
    const float* embed = (const float*)d_in[5];
    const float* We = (const float*)d_in[6];  const float* be = (const float*)d_in[7];
    const float* Wv = (const float*)d_in[8];  const float* bv = (const float*)d_in[9];
    const float* Wa = (const float*)d_in[10]; const float* ba = (const float*)d_in[11];
    const float* Wt = (const float*)d_in[12]; const float* bt = (const float*)d_in[13];
    const float* Wu = (const float*)d_in[14]; const float* bu = (const float*)d_in[15];

    // Workspace layout (all f32). e_bar..t_u contiguous for one-shot zeroing.
    float* hv    = (float*)d_ws;
    float* u     = hv    + (size_t)N_ATOMS * D;
    float* e_bar = u     + (size_t)N_MOLS  * D;
    float* e_u   = e_bar + (size_t)N_ATOMS * D;
    float* v_u   = e_u   + (size_t)N_MOLS  * D;
    float* a_u   = v_u   + (size_t)N_MOLS  * D;
    float* t_u   = a_u   + (size_t)N_MOLS  * D;

    // u starts at zero; h_v = embed[atoms]
    {
        const long n4 = (long)N_MOLS * D / 4;
        zero_kernel<<<div_up_l(n4, 256), 256, 0, stream>>>(u, n4);
    }
    init_hv_kernel<<<div_up_l((long)N_ATOMS * 32, 256), 256, 0, stream>>>(atoms, embed, hv);

    for (int it = 0; it < REPEAT; ++it) {
        // zero e_bar, e_u, v_u, a_u, t_u (contiguous)
        const long n4 = ((long)N_ATOMS * D + 4L * N_MOLS * D) / 4;
        zero_kernel<<<div_up_l(n4, 256), 256, 0, stream>>>(e_bar, n4);

        edge_kernel<<<div_up_l(N_BONDS, ROWS), 256, 0, stream>>>(
            hv, u, bond_idxs, atom_mol, We, be, e_bar, e_u);
        node_kernel<<<div_up_l(N_ATOMS, ROWS), 256, 0, stream>>>(
            hv, e_bar, u, atom_mol, Wv, bv, v_u);
        angle_kernel<<<div_up_l(N_ANGLES, ROWS), 256, 0, stream>>>(
            hv, u, angle_idxs, atom_mol, Wa, ba, a_u);
        torsion_kernel<<<div_up_l(N_TORSIONS, ROWS), 256, 0, stream>>>(
            hv, u, torsion_idxs, atom_mol, Wt, bt, t_u);
        u_kernel<<<div_up_l(N_MOLS, ROWS), 256, 0, stream>>>(
            u, e_u, v_u, a_u, t_u, Wu, bu);
    }

    {
        const long n4 = (long)N_MOLS * D / 4;
        copy_kernel<<<div_up_l(n4, 256), 256, 0, stream>>>((float*)d_out, u, n4);
    }
}